// ShareLayer_19207093747934
// MI455X (gfx1250) — compile-verified
//
#include <hip/hip_runtime.h>

// ShareLayer: warp(x, reg_in) -> group max/mean over chuck=4 -> warp(share, reg_out)
// B=8, C=32, H=W=256, chuck=4, groups G=2, shared channels C2=64.
// Memory-bound gather op: no matmul structure -> no WMMA; optimize for HBM/L2.

#define HWn   65536
#define Wn    256
#define Cn    32
#define C2n   64
#define CHKn  4
#define Gn    2

typedef float f4 __attribute__((ext_vector_type(4)));

struct Coord { int i00; int dx1; int dyW; float wx, wy; };

__device__ __forceinline__ Coord make_coord(float px, float py, float dx, float dy) {
    // exactly mirrors the reference formula (incl. clip order) in fp32
    float tx = px + dx;
    float gx = 2.0f * tx / 255.0f - 1.0f;
    gx = fminf(fmaxf(gx, -2.0f), 2.0f);
    float ix = (gx + 1.0f) * 0.5f * 255.0f;
    ix = fminf(fmaxf(ix, 0.0f), 255.0f);

    float ty = py + dy;
    float gy = 2.0f * ty / 255.0f - 1.0f;
    gy = fminf(fmaxf(gy, -2.0f), 2.0f);
    float iy = (gy + 1.0f) * 0.5f * 255.0f;
    iy = fminf(fmaxf(iy, 0.0f), 255.0f);

    float x0f = floorf(ix), y0f = floorf(iy);
    Coord c;
    c.wx = ix - x0f;
    c.wy = iy - y0f;
    int x0 = (int)x0f, y0 = (int)y0f;
    int x1 = min(x0 + 1, Wn - 1);
    int y1 = min(y0 + 1, Wn - 1);
    c.i00 = y0 * Wn + x0;
    c.dx1 = x1 - x0;
    c.dyW = (y1 - y0) * Wn;
    return c;
}

__device__ __forceinline__ float bilerp(const float* __restrict__ p, const Coord& c) {
    float v00 = p[c.i00];
    float v01 = p[c.i00 + c.dx1];
    float v10 = p[c.i00 + c.dyW];
    float v11 = p[c.i00 + c.dx1 + c.dyW];
    float w00 = (1.0f - c.wx) * (1.0f - c.wy);
    float w01 = c.wx * (1.0f - c.wy);
    float w10 = (1.0f - c.wx) * c.wy;
    float w11 = c.wx * c.wy;
    return v00 * w00 + v01 * w01 + v10 * w10 + v11 * w11;
}

// ---------------- Kernel 1: fused warp1 + group max/mean ----------------
// grid: (HW/(4*128), 32/8, G)  block: 128 threads, 4 pixels/thread, 8 channels/block.y
__global__ __launch_bounds__(128) void k_warp_reduce(
    const float* __restrict__ x, const float* __restrict__ reg, float* __restrict__ xs)
{
    const int t  = threadIdx.x;
    const int g  = blockIdx.z;
    const int c0 = blockIdx.y * 8;
    const int p0 = (blockIdx.x * 128 + t) * 4;      // 4 consecutive pixels, same row
    const float py = (float)(p0 >> 8);
    const int   w0 = p0 & (Wn - 1);

    // coordinates for 4 batches x 4 pixels (amortized over 8 channels)
    Coord cs[4][4];
    #pragma unroll
    for (int b = 0; b < 4; ++b) {
        const int bb = g * CHKn + b;
        const f4 dx4 = *(const f4*)(reg + (size_t)(bb * 4 + 0) * HWn + p0);
        const f4 dy4 = *(const f4*)(reg + (size_t)(bb * 4 + 1) * HWn + p0);
        #pragma unroll
        for (int j = 0; j < 4; ++j)
            cs[b][j] = make_coord((float)(w0 + j), py, dx4[j], dy4[j]);
    }

    float mx[8][4], sm[8][4];
    #pragma unroll
    for (int ci = 0; ci < 8; ++ci)
        #pragma unroll
        for (int j = 0; j < 4; ++j) { mx[ci][j] = -3.4028235e38f; sm[ci][j] = 0.0f; }

    for (int b = 0; b < 4; ++b) {                    // batch outer: reuse coords
        const size_t pbase = ((size_t)(g * CHKn + b) * Cn + c0) * HWn;
        #pragma unroll
        for (int ci = 0; ci < 8; ++ci) {
            const float* __restrict__ plane = x + pbase + (size_t)ci * HWn;
            #pragma unroll
            for (int j = 0; j < 4; ++j) {
                float v = bilerp(plane, cs[b][j]);
                mx[ci][j] = fmaxf(mx[ci][j], v);
                sm[ci][j] += v;
            }
        }
    }

    #pragma unroll
    for (int ci = 0; ci < 8; ++ci) {
        const int c = c0 + ci;
        f4 omax = { mx[ci][0], mx[ci][1], mx[ci][2], mx[ci][3] };
        f4 omean = { 0.25f * sm[ci][0], 0.25f * sm[ci][1],
                     0.25f * sm[ci][2], 0.25f * sm[ci][3] };
        // keep x_share regular-temporal: kernel 2 re-reads it from L2 (fits easily in 192MB)
        *(f4*)(xs + ((size_t)g * C2n + c) * HWn + p0) = omax;
        *(f4*)(xs + ((size_t)g * C2n + Cn + c) * HWn + p0) = omean;
    }
}

// ---------------- Kernel 2: warp2 over broadcast x_share ----------------
// grid: (HW/(4*256), 64/8, B)  block: 256 threads, 4 pixels/thread, 8 channels/block.y
__global__ __launch_bounds__(256) void k_warp_out(
    const float* __restrict__ xs, const float* __restrict__ reg, float* __restrict__ out)
{
    __shared__ __align__(16) float smem[2048];      // dx tile [1024] + dy tile [1024]
    const int t  = threadIdx.x;
    const int b  = blockIdx.z;
    const int g  = b >> 2;
    const int c0 = blockIdx.y * 8;
    const int p0 = (blockIdx.x * 256 + t) * 4;

    // Stage this block's reg_out tile through LDS with the CDNA5 async path
    // (ASYNCcnt). Each lane lands its own 16B, so s_wait_asynccnt suffices
    // (no cross-lane LDS sharing -> no barrier needed).
    const float* sdx = reg + (size_t)(b * 4 + 2) * HWn;   // reg_out dx plane
    const float* sdy = reg + (size_t)(b * 4 + 3) * HWn;   // reg_out dy plane
    unsigned voff = (unsigned)(p0 * 4);                   // byte offset within plane
    unsigned lds0 = (unsigned)(size_t)(&smem[t * 4]);
    unsigned lds1 = (unsigned)(size_t)(&smem[1024 + t * 4]);
    asm volatile(
        "global_load_async_to_lds_b128 %0, %1, %2\n\t"
        "global_load_async_to_lds_b128 %3, %1, %4\n\t"
        "s_wait_asynccnt 0"
        :
        : "v"(lds0), "v"(voff), "s"(sdx), "v"(lds1), "s"(sdy)
        : "memory");

    const f4 dx4 = *(const f4*)(&smem[t * 4]);
    const f4 dy4 = *(const f4*)(&smem[1024 + t * 4]);

    const float py = (float)(p0 >> 8);
    const int   w0 = p0 & (Wn - 1);
    Coord cs[4];
    #pragma unroll
    for (int j = 0; j < 4; ++j)
        cs[j] = make_coord((float)(w0 + j), py, dx4[j], dy4[j]);

    #pragma unroll
    for (int ci = 0; ci < 8; ++ci) {
        const int c = c0 + ci;
        const float* __restrict__ plane = xs + ((size_t)g * C2n + c) * HWn;
        f4 o;
        #pragma unroll
        for (int j = 0; j < 4; ++j) o[j] = bilerp(plane, cs[j]);
        // output is write-once: non-temporal store, don't evict x_share/x from L2
        __builtin_nontemporal_store(o, (f4*)(out + ((size_t)b * C2n + c) * HWn + p0));
    }
}

extern "C" void kernel_launch(void* const* d_in, const int* in_sizes, int n_in,
                              void* d_out, int out_size, void* d_ws, size_t ws_size,
                              hipStream_t stream) {
    (void)in_sizes; (void)n_in; (void)out_size; (void)ws_size;
    const float* x   = (const float*)d_in[0];   // [8,32,256,256] f32
    const float* reg = (const float*)d_in[1];   // [8,4,256,256]  f32
    // d_in[2] is chuck==4 (compile-time constant here; can't be read during capture)
    float* xs  = (float*)d_ws;                  // [2,64,256,256] f32 = 33.5 MB scratch
    float* out = (float*)d_out;                 // [8,64,256,256] f32

    // Kernel 1: fused warp+reduce. 128 blocks/pixels-dim x 4 chan-chunks x 2 groups
    k_warp_reduce<<<dim3(128, 4, 2), dim3(128), 0, stream>>>(x, reg, xs);
    // Kernel 2: final warp. 64 x 8 chan-chunks x 8 batches
    k_warp_out<<<dim3(64, 8, 8), dim3(256), 0, stream>>>(xs, reg, out);
}